// VectorizedMaskingStrategy_55267639164951
// MI455X (gfx1250) — compile-verified
//
#include <hip/hip_runtime.h>
#include <hip/hip_bf16.h>
#include <stdint.h>

// Problem constants (fixed by the reference): features [128, 16384, 6] fp32.
#define BATCH 128
#define SEQ 16384
#define FEAT 6
#define TARGET 2457            // int(16384 * 0.15)
#define NSPANS 393             // max(1, int(2457 / 12.5 * 2))
#define SPAN_MINV 5
#define SPAN_MAXV 20
#define MASK_WORDS_PER_ROW (SEQ / 32)   // 512 u32 per row

#define CHUNK_POS 1024
#define CHUNK_FLOATS (CHUNK_POS * FEAT)    // 6144 floats = 24 KB
#define PAIRS_PER_ROW 8                    // each block pipelines 2 chunks

// ---------------- Threefry-2x32 (JAX's PRNG family), 20 rounds ----------------
__device__ __forceinline__ void tfr(unsigned &x0, unsigned &x1, int r) {
    x0 += x1;
    x1 = (x1 << r) | (x1 >> (32 - r));
    x1 ^= x0;
}
struct U2 { unsigned x, y; };
__device__ __forceinline__ U2 threefry2x32(unsigned k0, unsigned k1,
                                           unsigned c0, unsigned c1) {
    unsigned k2 = 0x1BD11BDAu ^ k0 ^ k1;
    unsigned x0 = c0 + k0, x1 = c1 + k1;
    tfr(x0,x1,13); tfr(x0,x1,15); tfr(x0,x1,26); tfr(x0,x1,6);
    x0 += k1; x1 += k2 + 1u;
    tfr(x0,x1,17); tfr(x0,x1,29); tfr(x0,x1,16); tfr(x0,x1,24);
    x0 += k2; x1 += k0 + 2u;
    tfr(x0,x1,13); tfr(x0,x1,15); tfr(x0,x1,26); tfr(x0,x1,6);
    x0 += k0; x1 += k1 + 3u;
    tfr(x0,x1,17); tfr(x0,x1,29); tfr(x0,x1,16); tfr(x0,x1,24);
    x0 += k1; x1 += k2 + 4u;
    tfr(x0,x1,13); tfr(x0,x1,15); tfr(x0,x1,26); tfr(x0,x1,6);
    x0 += k2; x1 += k0 + 5u;
    return {x0, x1};
}

// =============================================================================
// Kernel 1: per-row span mask generation with exact-count top-k (radix select).
// One workgroup (8 wave32) per batch row. All selection work stays in LDS.
// =============================================================================
__global__ __launch_bounds__(256)
void maskgen_kernel(unsigned* __restrict__ maskWords) {
    const int b   = blockIdx.x;
    const int tid = threadIdx.x;

    __shared__ unsigned short score[SEQ];               // 32 KB: cov<<15 | noise15
    __shared__ unsigned cov[MASK_WORDS_PER_ROW];        // 2 KB coverage bitmask
    __shared__ unsigned hist[16];
    __shared__ unsigned scanbuf[256];
    __shared__ unsigned sh_prefix, sh_k;

    // Sub-keys derived from seed 1234 (key = (0, 1234), split via threefry).
    U2 kSpan  = threefry2x32(0u, 1234u, 0u, 0u);
    U2 kNoise = threefry2x32(0u, 1234u, 0u, 1u);

    for (int i = tid; i < MASK_WORDS_PER_ROW; i += 256) cov[i] = 0u;
    __syncthreads();

    // --- Span scatter: coverage via ds_or atomics (word-granular bit runs) --
    for (int i = tid; i < NSPANS; i += 256) {
        U2 r = threefry2x32(kSpan.x, kSpan.y, (unsigned)b, (unsigned)i);
        int len   = SPAN_MINV + (int)(r.x % (unsigned)(SPAN_MAXV - SPAN_MINV + 1));
        int start = (int)(r.y % (unsigned)(SEQ - SPAN_MINV));
        int end   = min(start + len, SEQ);
        int s = start;
        while (s < end) {
            int w = s >> 5, bit = s & 31;
            int take = min(32 - bit, end - s);
            unsigned m = (take == 32) ? 0xFFFFFFFFu : (((1u << take) - 1u) << bit);
            atomicOr(&cov[w], m);
            s += take;
        }
    }
    __syncthreads();

    // --- Scores: masked in (1,2), unmasked in (0,1), packed to u16 ----------
    const int base = tid * 64;                    // contiguous 64 positions/thread
    for (int j = 0; j < 64; ++j) {
        int s = base + j;
        U2 r = threefry2x32(kNoise.x, kNoise.y, (unsigned)b, (unsigned)s);
        unsigned c = (cov[s >> 5] >> (s & 31)) & 1u;
        score[s] = (unsigned short)((c << 15) | (r.x >> 17));
    }
    if (tid == 0) { sh_prefix = 0u; sh_k = TARGET; }
    __syncthreads();

    // --- Exact top-TARGET via 4x4-bit MSB-first radix select ----------------
    for (int pass = 0; pass < 4; ++pass) {
        const int shift = 12 - pass * 4;
        if (tid < 16) hist[tid] = 0u;
        __syncthreads();
        const unsigned prefix = sh_prefix;
        const unsigned hmask  = 0xFFFFu & ~((1u << (shift + 4)) - 1u);
        for (int j = 0; j < 64; ++j) {
            unsigned v = (unsigned)score[base + j];
            if ((v & hmask) == prefix) atomicAdd(&hist[(v >> shift) & 15u], 1u);
        }
        __syncthreads();
        if (tid == 0) {
            unsigned k = sh_k, c = 0; int d = 15;
            for (; d >= 0; --d) { c += hist[d]; if (c >= k) break; }
            sh_k      = k - (c - hist[d]);            // rank within tied bucket
            sh_prefix = prefix | ((unsigned)d << shift);
        }
        __syncthreads();
    }
    const unsigned T   = sh_prefix;   // exact threshold value (u16)
    const unsigned kEq = sh_k;        // how many score==T to take (>=1)

    // --- Deterministic tie break: lowest-index ties win ---------------------
    unsigned eq = 0;
    for (int j = 0; j < 64; ++j) if ((unsigned)score[base + j] == T) ++eq;
    scanbuf[tid] = eq;
    __syncthreads();
    if (tid == 0) {
        unsigned run = 0;
        for (int i = 0; i < 256; ++i) { unsigned t = scanbuf[i]; scanbuf[i] = run; run += t; }
    }
    __syncthreads();
    unsigned rank = scanbuf[tid];

    // --- Emit 64 mask bits per thread (2 words) to workspace ----------------
    unsigned w0 = 0, w1 = 0;
    for (int j = 0; j < 64; ++j) {
        unsigned v = (unsigned)score[base + j];
        bool m = false;
        if (v > T) m = true;
        else if (v == T) { m = (rank < kEq); ++rank; }
        if (m) { if (j < 32) w0 |= 1u << j; else w1 |= 1u << (j - 32); }
    }
    maskWords[(size_t)b * MASK_WORDS_PER_ROW + 2 * tid]     = w0;
    maskWords[(size_t)b * MASK_WORDS_PER_ROW + 2 * tid + 1] = w1;
}

// =============================================================================
// Kernel 2: streaming apply, double-buffered through the Tensor Data Mover.
//   - wave 0 issues two back-to-back TDM tile loads (24 KB each, TENSORcnt);
//     TDM ops from one wave complete in order, so s_wait_tensorcnt(1)
//     releases tile 0 while tile 1 is still streaming (software pipeline).
//   - the small irregular mask-word fetch uses global_load_async_to_lds_b128
//     (ASYNCcnt).
//   - compute: 6x ds_load_b128 per thread, cndmask select, 6x b128 stores.
// =============================================================================
typedef unsigned v4u_t __attribute__((ext_vector_type(4)));
typedef int      v8i_t __attribute__((ext_vector_type(8)));
typedef int      v4i_t __attribute__((ext_vector_type(4)));

// 1-D TDM descriptor: nElems 4-byte elements, global -> LDS.
__device__ __forceinline__ void tdm_load_1d(const void* gptr, unsigned ldsAddr,
                                            unsigned nElems) {
    unsigned long long g = (unsigned long long)gptr;
    v4u_t g0 = { 1u,                                    // count=1, user mode
                 ldsAddr,                               // lds_addr (bytes)
                 (unsigned)(g & 0xFFFFFFFFu),           // global_addr[31:0]
                 (unsigned)(((g >> 32) & 0x01FFFFFFu) | (2u << 30)) };  // type=2
    v8i_t g1 = { (int)(2u << 16),                       // data_size = 4 B
                 (int)(nElems << 16),                   // tensor_dim0 lo16
                 (int)((nElems >> 16) | (1u << 16)),    // dim0 hi16 | tensor_dim1=1
                 (int)(nElems << 16),                   // tile_dim0 (16-bit)
                 1,                                     // tile_dim1 = 1
                 (int)nElems,                           // tensor_dim0_stride
                 0, 0 };
    v4i_t gz = {0, 0, 0, 0};
#if __clang_major__ <= 22
    __builtin_amdgcn_tensor_load_to_lds(g0, g1, gz, gz, 0);
#else
    v8i_t gz8 = {0, 0, 0, 0, 0, 0, 0, 0};
    __builtin_amdgcn_tensor_load_to_lds(g0, g1, gz, gz, gz8, 0);
#endif
}

// Apply mask token over one staged 1024-position tile.
__device__ __forceinline__ void apply_chunk(const float* __restrict__ stagebuf,
                                            const unsigned* __restrict__ mwp,
                                            const float* __restrict__ mt,
                                            float* __restrict__ goutBase, int tid) {
    const float4* sp = (const float4*)(stagebuf + tid * 24);
    float vals[24];
#pragma unroll
    for (int q = 0; q < 6; ++q) {
        float4 t = sp[q];                                 // ds_load_b128
        vals[4*q+0] = t.x; vals[4*q+1] = t.y; vals[4*q+2] = t.z; vals[4*q+3] = t.w;
    }
    bool msk[4];
#pragma unroll
    for (int p = 0; p < 4; ++p) {
        unsigned pp  = (unsigned)(tid * 4 + p);
        unsigned bit = (mwp[pp >> 5] >> (pp & 31u)) & 1u;
        float f0 = vals[p * 6];                           // channel 0
        msk[p] = (bit != 0u) && (f0 == f0);               // mask & ~isnan
    }
    float res[24];
#pragma unroll
    for (int e = 0; e < 24; ++e) {
        int p = e / 6, ch = e % 6;
        res[e] = msk[p] ? mt[ch] : vals[e];
    }
    float4* gout = (float4*)(goutBase + (size_t)tid * 24);
#pragma unroll
    for (int q = 0; q < 6; ++q) {
        float4 t;
        t.x = res[4*q+0]; t.y = res[4*q+1]; t.z = res[4*q+2]; t.w = res[4*q+3];
        gout[q] = t;                                      // global_store_b128
    }
}

__global__ __launch_bounds__(256)
void apply_kernel(const float* __restrict__ feat,
                  const float* __restrict__ mtok,
                  const unsigned* __restrict__ maskWords,
                  float* __restrict__ out) {
    const int tid    = threadIdx.x;
    const int b      = blockIdx.x >> 3;          // 8 chunk-pairs per row
    const int chunk0 = (blockIdx.x & 7) * 2;     // this block: chunk0, chunk0+1

    __shared__ __align__(16) float stage[2][CHUNK_FLOATS];   // 2 x 24 KB tiles
    __shared__ __align__(16) unsigned mw[64];                // 2 x 32 mask words

    const size_t rowFloatBase = (size_t)b * (size_t)(SEQ * FEAT);
    const size_t c0FloatBase  = rowFloatBase + (size_t)chunk0 * CHUNK_FLOATS;

    // --- Wave 0: two pipelined TDM tile loads (TENSORcnt-tracked, in-order) -
    if (tid < 32) {
        tdm_load_1d(feat + c0FloatBase,
                    (unsigned)(size_t)(const void*)&stage[0][0], CHUNK_FLOATS);
        tdm_load_1d(feat + c0FloatBase + CHUNK_FLOATS,
                    (unsigned)(size_t)(const void*)&stage[1][0], CHUNK_FLOATS);
    }

    // --- Lanes 0-15 of wave 0: 256 B of mask words via async LDS load -------
    if (tid < 16) {
        const unsigned* msrc =
            maskWords + (size_t)b * MASK_WORDS_PER_ROW + chunk0 * 32 + tid * 4;
        unsigned ldsOff = (unsigned)(size_t)(const void*)&mw[0] + (unsigned)tid * 16u;
        unsigned long long gaddr = (unsigned long long)(const void*)msrc;
        asm volatile("global_load_async_to_lds_b128 %0, %1, off"
                     :: "v"(ldsOff), "v"(gaddr) : "memory");
    }
    asm volatile("s_wait_asynccnt 0x0" ::: "memory");
    if (tid < 32) __builtin_amdgcn_s_wait_tensorcnt(1);   // tile 0 landed
    __syncthreads();

    // Mask token: uniform address -> scalar-cached loads.
    float mt[FEAT];
#pragma unroll
    for (int f = 0; f < FEAT; ++f) mt[f] = mtok[f];

    // --- Process tile 0 while TDM streams tile 1 ----------------------------
    apply_chunk(&stage[0][0], &mw[0],  mt, out + c0FloatBase, tid);

    if (tid < 32) __builtin_amdgcn_s_wait_tensorcnt(0);   // tile 1 landed
    __syncthreads();

    // --- Process tile 1 -----------------------------------------------------
    apply_chunk(&stage[1][0], &mw[32], mt, out + c0FloatBase + CHUNK_FLOATS, tid);
}

// =============================================================================
extern "C" void kernel_launch(void* const* d_in, const int* in_sizes, int n_in,
                              void* d_out, int out_size, void* d_ws, size_t ws_size,
                              hipStream_t stream) {
    (void)in_sizes; (void)n_in; (void)out_size; (void)ws_size;
    const float* features   = (const float*)d_in[0];   // [128, 16384, 6] f32
    const float* mask_token = (const float*)d_in[1];   // [6] f32
    float* out = (float*)d_out;
    unsigned* maskWords = (unsigned*)d_ws;             // 128*512 u32 = 256 KB

    maskgen_kernel<<<BATCH, 256, 0, stream>>>(maskWords);
    apply_kernel<<<BATCH * PAIRS_PER_ROW, 256, 0, stream>>>(
        features, mask_token, maskWords, out);
}